// DVAT_5403068858731
// MI455X (gfx1250) — compile-verified
//
#include <hip/hip_runtime.h>
#include <hip/hip_bf16.h>
#include <cstdint>
#include <climits>

// Problem constants (from reference setup_inputs)
#define BB 4
#define SS 512
#define VV 30522
#define DD 768
#define KK 8
#define NSPECIAL 999

typedef __attribute__((ext_vector_type(2))) float v2f;   // f32 WMMA A/B: 16x4 / 4x16 -> 2 VGPRs
typedef __attribute__((ext_vector_type(8))) float v8f;   // f32 WMMA C/D: 16x16 -> 8 VGPRs

// ---------------------------------------------------------------------------
// Wave32 argmax reduction (value desc, index asc tie-break), butterfly shfl.
// ---------------------------------------------------------------------------
__device__ __forceinline__ void wave_argmax_vil(float &v, int &i, int &l) {
#pragma unroll
  for (int off = 16; off > 0; off >>= 1) {
    float ov = __shfl_xor(v, off, 32);
    int   oi = __shfl_xor(i, off, 32);
    int   ol = __shfl_xor(l, off, 32);
    if (ov > v || (ov == v && oi < i)) { v = ov; i = oi; l = ol; }
  }
}

__global__ __launch_bounds__(256) void adv_flip_kernel(
    const float* __restrict__ delta_grad,   // [B,S,D]
    const float* __restrict__ src_embeds,   // [B,S,D]
    const float* __restrict__ emb,          // [V,D]
    const int*   __restrict__ src_tokens,   // [B,S]
    const float* __restrict__ pred_lm,      // [B,S,V]
    const int*   __restrict__ attn_mask,    // [B,S]
    const float* __restrict__ rand_u,       // [B,S]
    int* __restrict__ out)                  // [B,S]
{
  const int pos  = blockIdx.x;              // flattened (b,s)
  const int tid  = threadIdx.x;
  const int lane = tid & 31;
  const int wid  = tid >> 5;
  const float NEG_INF = -__builtin_inff();

  __shared__ __align__(16) float s_rows[2 * DD];   // [delta row | src row], 6 KB
  __shared__ __align__(16) float s_emb[KK * DD];   // 8 candidate E rows, 24 KB
  __shared__ float s_wv[64];
  __shared__ int   s_wi[64];
  __shared__ int   s_top[KK];

  // -------------------------------------------------------------------------
  // Stage 0: async-copy this position's delta_grad/src_embeds rows (6 KB)
  // into LDS. Tracked by ASYNCcnt; overlaps with the 122 KB top-k scan below.
  // (low 32 bits of a flat pointer to LDS are the LDS byte offset)
  // -------------------------------------------------------------------------
  {
    const float* g0 = delta_grad + (size_t)pos * DD;
    const float* g1 = src_embeds + (size_t)pos * DD;
    unsigned lds0 = (unsigned)(size_t)(&s_rows[0]);
    // 2*768 floats = 384 x 16B chunks; 256 threads -> chunks tid, tid+256
    for (int c = tid; c < 2 * DD / 4; c += 256) {
      const float* g = (c < DD / 4) ? (g0 + c * 4) : (g1 + (c - DD / 4) * 4);
      unsigned dst = lds0 + (unsigned)c * 16u;
      unsigned long long ga = (unsigned long long)(const void*)g;
      asm volatile("global_load_async_to_lds_b128 %0, %1, off"
                   :: "v"(dst), "v"(ga) : "memory");
    }
  }

  // -------------------------------------------------------------------------
  // Stage 1: per-thread sorted top-8 over the pred_lm row (coalesced b64).
  // Row byte offset is 8-aligned (V*4 = 122088), so float2 is safe.
  // -------------------------------------------------------------------------
  const float2* row2 = (const float2*)(pred_lm + (size_t)pos * VV);
  const int N2 = VV / 2;  // 15261
  float tv[KK]; int ti[KK];
#pragma unroll
  for (int k = 0; k < KK; ++k) { tv[k] = NEG_INF; ti[k] = INT_MAX; }

  for (int i2 = tid; i2 < N2; i2 += 256) {
    if ((lane & 15) == 0)  // 2 prefetches per wave-iter: cover the +16KB stream
      __builtin_prefetch((const void*)(row2 + i2 + 256 * 8), 0, 0);
    float2 p = row2[i2];
    int base = i2 * 2;
#pragma unroll
    for (int j = 0; j < 2; ++j) {
      float v  = (j == 0) ? p.x : p.y;
      int  idx = base + j;
      if (v > tv[KK - 1]) {               // strict > keeps earlier index on tie
        tv[KK - 1] = v; ti[KK - 1] = idx;
#pragma unroll
        for (int k = KK - 1; k > 0; --k) { // one bubble pass restores order
          bool sw = (tv[k] > tv[k - 1]) ||
                    (tv[k] == tv[k - 1] && ti[k] < ti[k - 1]);
          if (sw) {
            float fv = tv[k]; tv[k] = tv[k - 1]; tv[k - 1] = fv;
            int   fi = ti[k]; ti[k] = ti[k - 1]; ti[k - 1] = fi;
          }
        }
      }
    }
  }

  // -------------------------------------------------------------------------
  // Stage 2a: merge 32 sorted lists -> wave top-8 (head-pointer + wave argmax)
  // -------------------------------------------------------------------------
  {
    int h = 0;
    for (int r = 0; r < KK; ++r) {
      float v = (h < KK) ? tv[h] : NEG_INF;
      int   i = (h < KK) ? ti[h] : INT_MAX;
      int  wl = lane;
      wave_argmax_vil(v, i, wl);
      if (lane == wl) ++h;
      if (lane == 0) { s_wv[wid * KK + r] = v; s_wi[wid * KK + r] = i; }
    }
  }
  __syncthreads();

  // Stage 2b: wave 0 merges the 8 wave-lists (64 entries, 2 per lane)
  if (wid == 0) {
    float a_v = s_wv[lane], b_v = s_wv[lane + 32];
    int   a_i = s_wi[lane], b_i = s_wi[lane + 32];
    if (b_v > a_v || (b_v == a_v && b_i < a_i)) {
      float t = a_v; a_v = b_v; b_v = t;
      int   u = a_i; a_i = b_i; b_i = u;
    }
    int h = 0;
    for (int r = 0; r < KK; ++r) {
      float v = (h == 0) ? a_v : (h == 1) ? b_v : NEG_INF;
      int   i = (h == 0) ? a_i : (h == 1) ? b_i : INT_MAX;
      int  wl = lane;
      wave_argmax_vil(v, i, wl);
      if (lane == wl) ++h;
      if (lane == 0) s_top[r] = i;
    }
  }
  __syncthreads();   // publish s_top to all waves

  const int attn = attn_mask[pos];

  // -------------------------------------------------------------------------
  // Stage 2c: gather the 8 candidate embedding rows into LDS in ONE async
  // burst (wave w copies E[c_w], 6 x b128 per lane) instead of 192 serialized
  // per-chunk gathers inside the WMMA loop.
  // -------------------------------------------------------------------------
  {
    int cw = s_top[wid];
    if (attn == 0) cw = 0;
    const float* gsrc = emb + (size_t)cw * DD;
    unsigned ldsdst = (unsigned)(size_t)(&s_emb[wid * DD]);
#pragma unroll
    for (int j = 0; j < 6; ++j) {
      int chunk = lane + 32 * j;                     // 192 x 16B = 3 KB row
      unsigned dst = ldsdst + (unsigned)chunk * 16u;
      unsigned long long ga = (unsigned long long)(const void*)(gsrc + chunk * 4);
      asm volatile("global_load_async_to_lds_b128 %0, %1, off"
                   :: "v"(dst), "v"(ga) : "memory");
    }
  }

  // Every wave waits for ITS OWN async LDS writes (s_rows batch + s_emb
  // batch), then the barrier makes all waves' writes visible to wave 0.
  asm volatile("s_wait_asynccnt 0x0" ::: "memory");
  __syncthreads();

  // -------------------------------------------------------------------------
  // Stage 3 (wave 0, full EXEC): one V_WMMA_F32_16X16X4_F32 accumulation
  // chain computes ALL cross products at f32 precision:
  //   A row0 = delta_grad row, A row1 = src row (rows 2-15 zero)
  //   B cols 0-7 = E[c_0..7] (from s_emb), col 8 = src row
  //   => D[0][N] = dg.E[cN], D[1][N] = src.E[cN], D[0][8] = dg.src,
  //      D[1][8] = src.src.   ||E[cN]||^2 accumulated locally from the B
  //      operand registers (lane N / N+16 hold E[cN]'s two K-halves).
  // All operands come from LDS via a per-lane base pointer -> branch-free
  // loop body: 2x ds_load_b64 + 4x cndmask + 2x fmac + 1x wmma.
  // ISA layout: 32-bit A 16x4 / B 4x16 -> lane-half selects K-half,
  // VGPR selects K within the half; C/D VGPR r = rows r / r+8 by lane half.
  // -------------------------------------------------------------------------
  if (wid == 0) {
    const int stok = src_tokens[pos];

    const int N    = lane & 15;            // output column this lane feeds
    const int koff = (lane >> 4) << 1;     // K-half: 0 or 2

    int c = 0;
    if (N < KK) { c = s_top[N]; if (attn == 0) c = 0; }  // topk_idx * mask

    const bool avalid = (N < 2);
    const bool bvalid = (N <= KK);
    const float* abase = &s_rows[(N == 0) ? 0 : DD];     // dg (M=0) / src (M=1)
    const float* bbase = (N < KK)  ? &s_emb[N * DD]      // E[cN]
                       : (N == KK) ? &s_rows[DD]         // src row for col 8
                                   : &s_rows[0];         // dummy (zeroed)

    v8f  acc = {};
    float ee = 0.0f;
#pragma unroll 4
    for (int k0 = 0; k0 < DD; k0 += 4) {
      float2 al = *(const float2*)(abase + k0 + koff);   // ds_load_b64
      float2 bl = *(const float2*)(bbase + k0 + koff);   // ds_load_b64
      v2f a; a.x = avalid ? al.x : 0.0f; a.y = avalid ? al.y : 0.0f;
      v2f b; b.x = bvalid ? bl.x : 0.0f; b.y = bvalid ? bl.y : 0.0f;
      ee = fmaf(b.x, b.x, ee);             // local ||E[cN]||^2 partial
      ee = fmaf(b.y, b.y, ee);             // (lanes N>=8: unused / zero)
      acc = __builtin_amdgcn_wmma_f32_16x16x4_f32(
                /*neg_a=*/false, a, /*neg_b=*/false, b,
                /*c_mod=*/(short)0, acc, /*reuse_a=*/false, /*reuse_b=*/false);
    }

    ee += __shfl_xor(ee, 16, 32);          // combine the two K-halves

    float dgE = acc[0];                    // D[0][lane]  (lanes 0-15)
    float sEv = acc[1];                    // D[1][lane]
    float pd  = __shfl(dgE, 8, 32);        // D[0][8] = dg.src
    float ssq = __shfl(sEv, 8, 32);        // D[1][8] = src.src

    float score = NEG_INF;
    int   ctok  = c;
    if (lane < KK) {
      if (!(c < NSPECIAL || c == stok)) {  // special / original-token filters
        float sq = ssq + ee - 2.0f * sEv;             // ||src - E[c]||^2
        float dn = sqrtf(fmaxf(sq, 0.0f) + 1e-20f);   // dir_norm
        score = (dgE - pd) / dn;                      // dir_dot_grad
      }
    }
    // argmax over lanes 0-7 (xor group is closed, no contamination from 8-31)
#pragma unroll
    for (int off = 4; off > 0; off >>= 1) {
      float ov = __shfl_xor(score, off, 32);
      int   oc = __shfl_xor(ctok,  off, 32);
      if (ov > score || (ov == score && oc < ctok)) { score = ov; ctok = oc; }
    }
    if (lane == 0) {
      int adv = (score == NEG_INF) ? 0 : ctok;   // jnp.argmax(all -inf) -> 0
      bool no_special = (stok >= NSPECIAL);
      bool do_swap    = (rand_u[pos] > 0.7f);    // rand_u > (1 - SWAP_RATIO)
      out[pos] = (no_special && do_swap) ? adv : stok;
    }
  }
}

// ---------------------------------------------------------------------------
// Host launch: one block per (b,s) position.
// ---------------------------------------------------------------------------
extern "C" void kernel_launch(void* const* d_in, const int* in_sizes, int n_in,
                              void* d_out, int out_size, void* d_ws, size_t ws_size,
                              hipStream_t stream) {
  (void)in_sizes; (void)n_in; (void)d_ws; (void)ws_size; (void)out_size;
  const float* delta_grad = (const float*)d_in[0];
  const float* src_embeds = (const float*)d_in[1];
  const float* emb        = (const float*)d_in[2];
  const int*   src_tokens = (const int*)d_in[3];
  const float* pred_lm    = (const float*)d_in[4];
  const int*   attn_mask  = (const int*)d_in[5];
  const float* rand_u     = (const float*)d_in[6];
  int* out = (int*)d_out;

  dim3 grid(BB * SS);   // 2048 positions
  dim3 block(256);      // 8 wave32
  adv_flip_kernel<<<grid, block, 0, stream>>>(
      delta_grad, src_embeds, emb, src_tokens, pred_lm, attn_mask, rand_u, out);
}